// DglHGTFFDConvBlock_39367670235357
// MI455X (gfx1250) — compile-verified
//
#include <hip/hip_runtime.h>
#include <hip/hip_bf16.h>

// ---------------------------------------------------------------------------
// HGT conv block for MI455X (gfx1250, wave32, WMMA f32 16x16x4).
// N=100000 nodes/type, E=500000 edges/rel, IN=OUT=128, H=8, DK=16, DFF=512.
// Workspace budget: 99072 + 6*N*128 + E*8 + 2*N*8 floats ~= 330 MB.
// ---------------------------------------------------------------------------

typedef __attribute__((ext_vector_type(2))) float v2f;
typedef __attribute__((ext_vector_type(8))) float v8f;

// ---------------- helpers: ordered-uint encoding for float atomicMax -------
__device__ __forceinline__ unsigned orderedF(float f) {
    unsigned u = __float_as_uint(f);
    return (u & 0x80000000u) ? ~u : (u | 0x80000000u);
}
__device__ __forceinline__ float fromOrderedF(unsigned s) {
    unsigned u = (s & 0x80000000u) ? (s & 0x7FFFFFFFu) : ~s;
    return __uint_as_float(u);
}

// ---------------- zero fill -------------------------------------------------
__global__ __launch_bounds__(256) void zero_kernel(float* __restrict__ p, long n) {
    long i = (long)blockIdx.x * blockDim.x + threadIdx.x;
    long stride = (long)gridDim.x * blockDim.x;
    for (; i < n; i += stride) p[i] = 0.f;
}

// ---------------- fused relation weights -----------------------------------
// WF[r][i][h*16+j] = sum_d W[t(r)][i][h*16+d] * R[r][h][d][j]
// bF[r][h*16+j]    = sum_d b[t(r)][h*16+d]   * R[r][h][d][j]
// t(r) = src node type of relation r: {0,1,0}
__global__ __launch_bounds__(256)
void fuse_kernel(const float* __restrict__ W,   // [2,128,128]
                 const float* __restrict__ b,   // [2,128]
                 const float* __restrict__ R,   // [3,8,16,16]
                 float* __restrict__ WF,        // [3,128,128]
                 float* __restrict__ bF)        // [3,128]
{
    const int r = blockIdx.x;                   // 0..2
    const int t = (r == 1) ? 1 : 0;
    const float* Wt = W + (long)t * 16384;
    const float* bt = b + (long)t * 128;
    const float* Rr = R + (long)r * 2048;
    for (int idx = threadIdx.x; idx < 16384; idx += blockDim.x) {
        const int i = idx >> 7;
        const int hj = idx & 127;
        const int h = hj >> 4, j = hj & 15;
        float acc = 0.f;
#pragma unroll
        for (int d = 0; d < 16; ++d)
            acc += Wt[i * 128 + h * 16 + d] * Rr[h * 256 + d * 16 + j];
        WF[(long)r * 16384 + idx] = acc;
    }
    for (int hj = threadIdx.x; hj < 128; hj += blockDim.x) {
        const int h = hj >> 4, j = hj & 15;
        float acc = 0.f;
#pragma unroll
        for (int d = 0; d < 16; ++d)
            acc += bt[h * 16 + d] * Rr[h * 256 + d * 16 + j];
        bF[(long)r * 128 + hj] = acc;
    }
}

// ---------------- generic [rows,128]@[128,128] GEMM via WMMA f32 ------------
// C = op(A) @ B + bias (+ resid).  rows must be a multiple of 16.
// 64 output rows per block (4 row-tiles), 8 waves cover all 128 columns.
// Each wave keeps 4 independent WMMA accumulator chains and shares each B
// fragment LDS read across the 4 row-tiles.
template <bool RELU_A, bool RESID>
__global__ __launch_bounds__(256)
void gemm128_kernel(const float* __restrict__ A, const float* __restrict__ B,
                    const float* __restrict__ bias, const float* __restrict__ resid,
                    float* __restrict__ C, int rows)
{
    __shared__ float As[64][132];   // pad: K-split halves hit disjoint banks
    __shared__ float Bs[128][136];
    const int tid = threadIdx.x;
    const int wave = tid >> 5, lane = tid & 31;
    const long rowBase = (long)blockIdx.x * 64;

    // stage B (128x128) into LDS
    for (int i = tid; i < 128 * 32; i += 256) {
        const int rr = i >> 5, c4 = (i & 31) << 2;
        const float4 v = *(const float4*)(B + (long)rr * 128 + c4);
        Bs[rr][c4] = v.x; Bs[rr][c4 + 1] = v.y; Bs[rr][c4 + 2] = v.z; Bs[rr][c4 + 3] = v.w;
    }
    // stage A strip (64x128) into LDS (optional relu, zero-pad OOB rows)
    for (int i = tid; i < 64 * 32; i += 256) {
        const int rr = i >> 5, c4 = (i & 31) << 2;
        const long gr = rowBase + rr;
        float4 v = make_float4(0.f, 0.f, 0.f, 0.f);
        if (gr < rows) v = *(const float4*)(A + gr * 128 + c4);
        if (RELU_A) {
            v.x = fmaxf(v.x, 0.f); v.y = fmaxf(v.y, 0.f);
            v.z = fmaxf(v.z, 0.f); v.w = fmaxf(v.w, 0.f);
        }
        As[rr][c4] = v.x; As[rr][c4 + 1] = v.y; As[rr][c4 + 2] = v.z; As[rr][c4 + 3] = v.w;
    }
    __syncthreads();

    const int colBase = wave << 4;
    const int hi = lane >> 4;     // K half-select
    const int mn = lane & 15;     // A row / B,C,D column within tile
    v8f cacc[4];
    {
        const float bvv = bias[colBase + mn];
#pragma unroll
        for (int rt = 0; rt < 4; ++rt)
#pragma unroll
            for (int i = 0; i < 8; ++i) cacc[rt][i] = bvv;
    }
#pragma unroll
    for (int k0 = 0; k0 < 128; k0 += 4) {
        const int ka = k0 + (hi << 1);
        v2f bfrag;
        bfrag.x = Bs[ka][colBase + mn];
        bfrag.y = Bs[ka + 1][colBase + mn];
#pragma unroll
        for (int rt = 0; rt < 4; ++rt) {
            v2f a;
            a.x = As[rt * 16 + mn][ka];
            a.y = As[rt * 16 + mn][ka + 1];
            cacc[rt] = __builtin_amdgcn_wmma_f32_16x16x4_f32(false, a, false, bfrag,
                                                             (short)0, cacc[rt], false, false);
        }
    }
#pragma unroll
    for (int rt = 0; rt < 4; ++rt) {
        const long tileRow = rowBase + rt * 16;
        if (tileRow < rows) {
#pragma unroll
            for (int i = 0; i < 8; ++i) {
                const long r = tileRow + i + (hi << 3);
                float val = cacc[rt][i];
                if (RESID) val += resid[r * 128 + colBase + mn];
                C[r * 128 + colBase + mn] = val;
            }
        }
    }
}

// ---------------- edge scores + segment max ---------------------------------
// One wave per edge. lane l covers elements 4l..4l+3 of [H=8,DK=16] (head = l>>2).
__global__ __launch_bounds__(256)
void score_kernel(const float* __restrict__ Q, const float* __restrict__ KP,
                  const int* __restrict__ src, const int* __restrict__ dst,
                  const float* __restrict__ pri,   // 8 floats for this relation
                  float* __restrict__ sbuf,        // [E,8]
                  unsigned* __restrict__ mbuf,     // [N,8] ordered-uint
                  int E)
{
    const int e = blockIdx.x * 8 + (threadIdx.x >> 5);
    const int lane = threadIdx.x & 31;
    if (e >= E) return;
    const int s = src[e], d = dst[e];
    const float4 qv = *(const float4*)(Q + (long)d * 128 + lane * 4);
    const float4 kv = *(const float4*)(KP + (long)s * 128 + lane * 4);
    float p = qv.x * kv.x + qv.y * kv.y + qv.z * kv.z + qv.w * kv.w;
    p += __shfl_xor(p, 1, 32);
    p += __shfl_xor(p, 2, 32);
    const int h = lane >> 2;
    if ((lane & 3) == 0) {
        const float sc = p * pri[h] * 0.25f;   // 1/sqrt(DK), DK=16
        sbuf[(long)e * 8 + h] = sc;
        atomicMax(&mbuf[(long)d * 8 + h], orderedF(sc));
    }
}

// ---------------- exp(s - m) and segment sum --------------------------------
__global__ __launch_bounds__(256)
void expsum_kernel(float* __restrict__ sbuf, const int* __restrict__ dst,
                   const unsigned* __restrict__ mbuf, float* __restrict__ zbuf, int E)
{
    const long i = (long)blockIdx.x * 256 + threadIdx.x;   // over E*8
    if (i >= (long)E * 8) return;
    const int e = (int)(i >> 3);
    const int h = (int)(i & 7);
    const int d = dst[e];
    const float m = fromOrderedF(mbuf[(long)d * 8 + h]);
    const float ev = expf(sbuf[i] - m);
    sbuf[i] = ev;
    atomicAdd(&zbuf[(long)d * 8 + h], ev);
}

// ---------------- weighted aggregation into t[dst] --------------------------
__global__ __launch_bounds__(256)
void agg_kernel(const float* __restrict__ sbuf, const float* __restrict__ zbuf,
                const float* __restrict__ VP, const int* __restrict__ src,
                const int* __restrict__ dst, float* __restrict__ T, int E)
{
    const int e = blockIdx.x * 8 + (threadIdx.x >> 5);
    const int lane = threadIdx.x & 31;
    if (e >= E) return;
    const int s = src[e], d = dst[e];
    const int h = lane >> 2;
    const float a = sbuf[(long)e * 8 + h] / zbuf[(long)d * 8 + h];
    const float4 v = *(const float4*)(VP + (long)s * 128 + lane * 4);
    float* tp = T + (long)d * 128 + lane * 4;
    atomicAdd(tp + 0, v.x * a);
    atomicAdd(tp + 1, v.y * a);
    atomicAdd(tp + 2, v.z * a);
    atomicAdd(tp + 3, v.w * a);
}

// ---------------- row-wise LayerNorm ---------------------------------------
__global__ __launch_bounds__(256)
void ln_kernel(const float* __restrict__ Y, const float* __restrict__ g,
               const float* __restrict__ be, float* __restrict__ X, int rows)
{
    const int row = blockIdx.x * 8 + (threadIdx.x >> 5);
    const int lane = threadIdx.x & 31;
    if (row >= rows) return;
    const float4 v = *(const float4*)(Y + (long)row * 128 + lane * 4);
    float s = v.x + v.y + v.z + v.w;
    float ss = v.x * v.x + v.y * v.y + v.z * v.z + v.w * v.w;
#pragma unroll
    for (int o = 16; o > 0; o >>= 1) {
        s += __shfl_xor(s, o, 32);
        ss += __shfl_xor(ss, o, 32);
    }
    const float mu = s * (1.f / 128.f);
    const float var = ss * (1.f / 128.f) - mu * mu;
    const float inv = rsqrtf(var + 1e-5f);
    const float4 gv = *(const float4*)(g + lane * 4);
    const float4 bv = *(const float4*)(be + lane * 4);
    float4 o;
    o.x = (v.x - mu) * inv * gv.x + bv.x;
    o.y = (v.y - mu) * inv * gv.y + bv.y;
    o.z = (v.z - mu) * inv * gv.z + bv.z;
    o.w = (v.w - mu) * inv * gv.w + bv.w;
    *(float4*)(X + (long)row * 128 + lane * 4) = o;
}

// ---------------- fused FFN: relu(X@W1+b1)@W2+b2 ----------------------------
// 16-row strip per block; 16x512 hidden tile lives in LDS between the stages.
// Stage 1 interleaves the 4 column tiles of each wave: one A-fragment LDS read
// feeds 4 independent WMMA chains per k-step.
__global__ __launch_bounds__(256)
void ffn_kernel(const float* __restrict__ X, const float* __restrict__ W1,
                const float* __restrict__ b1, const float* __restrict__ W2,
                const float* __restrict__ b2, float* __restrict__ Out, int rows)
{
    (void)rows;
    __shared__ float Xs[16][132];
    __shared__ float Hs[16][520];
    const int tid = threadIdx.x, wave = tid >> 5, lane = tid & 31;
    const long rowBase = (long)blockIdx.x * 16;
    for (int i = tid; i < 16 * 32; i += 256) {
        const int rr = i >> 5, c4 = (i & 31) << 2;
        const float4 v = *(const float4*)(X + (rowBase + rr) * 128 + c4);
        Xs[rr][c4] = v.x; Xs[rr][c4 + 1] = v.y; Xs[rr][c4 + 2] = v.z; Xs[rr][c4 + 3] = v.w;
    }
    __syncthreads();
    const int hi = lane >> 4;
    const int mn = lane & 15;
    // stage 1: hidden = relu(X@W1+b1); wave owns 4 of 32 column tiles, interleaved
    {
        v8f cacc[4];
#pragma unroll
        for (int t = 0; t < 4; ++t) {
            const float bvv = b1[(((wave << 2) + t) << 4) + mn];
#pragma unroll
            for (int i = 0; i < 8; ++i) cacc[t][i] = bvv;
        }
#pragma unroll
        for (int k0 = 0; k0 < 128; k0 += 4) {
            const int ka = k0 + (hi << 1);
            v2f a;
            a.x = Xs[mn][ka]; a.y = Xs[mn][ka + 1];
#pragma unroll
            for (int t = 0; t < 4; ++t) {
                const int colBase = ((wave << 2) + t) << 4;
                v2f bfrag;
                bfrag.x = W1[(long)ka * 512 + colBase + mn];
                bfrag.y = W1[(long)(ka + 1) * 512 + colBase + mn];
                cacc[t] = __builtin_amdgcn_wmma_f32_16x16x4_f32(false, a, false, bfrag,
                                                                (short)0, cacc[t], false, false);
            }
        }
#pragma unroll
        for (int t = 0; t < 4; ++t) {
            const int colBase = ((wave << 2) + t) << 4;
#pragma unroll
            for (int i = 0; i < 8; ++i)
                Hs[i + (hi << 3)][colBase + mn] = fmaxf(cacc[t][i], 0.f);
        }
    }
    __syncthreads();
    // stage 2: out = hidden@W2 + b2; one column tile per wave
    {
        const int colBase = wave << 4;
        v8f c;
        const float bvv = b2[colBase + mn];
#pragma unroll
        for (int i = 0; i < 8; ++i) c[i] = bvv;
#pragma unroll 16
        for (int k0 = 0; k0 < 512; k0 += 4) {
            const int ka = k0 + (hi << 1);
            v2f a, bfrag;
            a.x = Hs[mn][ka]; a.y = Hs[mn][ka + 1];
            bfrag.x = W2[(long)ka * 128 + colBase + mn];
            bfrag.y = W2[(long)(ka + 1) * 128 + colBase + mn];
            c = __builtin_amdgcn_wmma_f32_16x16x4_f32(false, a, false, bfrag,
                                                      (short)0, c, false, false);
        }
#pragma unroll
        for (int i = 0; i < 8; ++i)
            Out[(rowBase + i + (hi << 3)) * 128 + colBase + mn] = c[i];
    }
}

// ---------------------------------------------------------------------------
extern "C" void kernel_launch(void* const* d_in, const int* in_sizes, int n_in,
                              void* d_out, int out_size, void* d_ws, size_t ws_size,
                              hipStream_t stream)
{
    (void)n_in; (void)out_size; (void)ws_size;
    const float* h_a = (const float*)d_in[0];
    const float* h_b = (const float*)d_in[1];
    const float* Wk = (const float*)d_in[2];
    const float* bk = (const float*)d_in[3];
    const float* Wq = (const float*)d_in[4];
    const float* bq = (const float*)d_in[5];
    const float* Wv = (const float*)d_in[6];
    const float* bv = (const float*)d_in[7];
    const float* Wa = (const float*)d_in[8];
    const float* ba = (const float*)d_in[9];
    const float* gamma = (const float*)d_in[10];
    const float* beta = (const float*)d_in[11];
    const float* W1 = (const float*)d_in[12];
    const float* b1 = (const float*)d_in[13];
    const float* W2 = (const float*)d_in[14];
    const float* b2 = (const float*)d_in[15];
    const float* rel_pri = (const float*)d_in[16];
    const float* rel_att = (const float*)d_in[17];
    const float* rel_msg = (const float*)d_in[18];
    const int* srcs[3] = {(const int*)d_in[19], (const int*)d_in[21], (const int*)d_in[23]};
    const int* dsts[3] = {(const int*)d_in[20], (const int*)d_in[22], (const int*)d_in[24]};
    const int Es[3] = {in_sizes[19], in_sizes[21], in_sizes[23]};

    const int N = in_sizes[0] / 128;
    float* ws = (float*)d_ws;

    // workspace layout (floats)
    float* WkF = ws;                         // 3*16384
    float* WvF = ws + 3 * 16384;             // 3*16384
    float* bkF = ws + 6 * 16384;             // 3*128
    float* bvF = bkF + 3 * 128;              // 3*128
    size_t off = 6 * 16384 + 6 * 128;        // 99072
    float* Qa = ws + off; off += (size_t)N * 128;
    float* Qb = ws + off; off += (size_t)N * 128;
    float* KP = ws + off; off += (size_t)N * 128;
    float* VP = ws + off; off += (size_t)N * 128;
    float* Ta = ws + off; off += (size_t)N * 128;
    float* Tb = ws + off; off += (size_t)N * 128;
    int Emax = Es[0]; if (Es[1] > Emax) Emax = Es[1]; if (Es[2] > Emax) Emax = Es[2];
    float* sbuf = ws + off; off += (size_t)Emax * 8;
    unsigned* mbuf = (unsigned*)(ws + off); off += (size_t)N * 8;
    float* zbuf = ws + off; off += (size_t)N * 8;

    const int gemmBlocks = (N + 63) / 64;    // 64 rows per block
    const int ffnBlocks = N / 16;            // N = 100000 -> 6250 exact

    // 1) fused relation projections (Wk*rel_att, Wv*rel_msg)
    fuse_kernel<<<3, 256, 0, stream>>>(Wk, bk, rel_att, WkF, bkF);
    fuse_kernel<<<3, 256, 0, stream>>>(Wv, bv, rel_msg, WvF, bvF);

    // 2) query projections per node type
    gemm128_kernel<false, false><<<gemmBlocks, 256, 0, stream>>>(h_a, Wq, bq, nullptr, Qa, N);
    gemm128_kernel<false, false><<<gemmBlocks, 256, 0, stream>>>(h_b, Wq + 16384, bq + 128, nullptr, Qb, N);

    // 3) zero accumulators
    zero_kernel<<<2048, 256, 0, stream>>>(Ta, (long)N * 128);
    zero_kernel<<<2048, 256, 0, stream>>>(Tb, (long)N * 128);

    // 4) relations: r0 a->b, r1 b->a, r2 a->a
    const float* srcH[3] = {h_a, h_b, h_a};
    const float* Qr[3] = {Qb, Qa, Qa};
    float* Tr[3] = {Tb, Ta, Ta};
    for (int r = 0; r < 3; ++r) {
        gemm128_kernel<false, false><<<gemmBlocks, 256, 0, stream>>>(
            srcH[r], WkF + (long)r * 16384, bkF + (long)r * 128, nullptr, KP, N);
        gemm128_kernel<false, false><<<gemmBlocks, 256, 0, stream>>>(
            srcH[r], WvF + (long)r * 16384, bvF + (long)r * 128, nullptr, VP, N);
        zero_kernel<<<1024, 256, 0, stream>>>((float*)mbuf, (long)N * 8);
        zero_kernel<<<1024, 256, 0, stream>>>(zbuf, (long)N * 8);
        const int E = Es[r];
        score_kernel<<<(E + 7) / 8, 256, 0, stream>>>(Qr[r], KP, srcs[r], dsts[r],
                                                      rel_pri + (long)r * 8, sbuf, mbuf, E);
        expsum_kernel<<<(int)(((long)E * 8 + 255) / 256), 256, 0, stream>>>(sbuf, dsts[r], mbuf, zbuf, E);
        agg_kernel<<<(E + 7) / 8, 256, 0, stream>>>(sbuf, zbuf, VP, srcs[r], dsts[r], Tr[r], E);
    }

    // 5) update: y = relu(t)@Wa + ba + h  (reuse KP/VP as y buffers)
    float* Ya = KP; float* Yb = VP;
    gemm128_kernel<true, true><<<gemmBlocks, 256, 0, stream>>>(Ta, Wa, ba, h_a, Ya, N);
    gemm128_kernel<true, true><<<gemmBlocks, 256, 0, stream>>>(Tb, Wa + 16384, ba + 128, h_b, Yb, N);

    // 6) layernorm (reuse Qa/Qb as x buffers)
    float* Xa = Qa; float* Xb = Qb;
    ln_kernel<<<(N + 7) / 8, 256, 0, stream>>>(Ya, gamma, beta, Xa, N);
    ln_kernel<<<(N + 7) / 8, 256, 0, stream>>>(Yb, gamma + 128, beta + 128, Xb, N);

    // 7) FFN into stacked output [2,N,128]
    float* outf = (float*)d_out;
    ffn_kernel<<<ffnBlocks, 256, 0, stream>>>(Xa, W1, b1, W2, b2, outf, N);
    ffn_kernel<<<ffnBlocks, 256, 0, stream>>>(Xb, W1 + 128 * 512, b1 + 512,
                                              W2 + 512 * 128, b2 + 128,
                                              outf + (size_t)N * 128, N);
}